// QInputEncoder_75033078661560
// MI455X (gfx1250) — compile-verified
//
#include <hip/hip_runtime.h>

// ---------- problem constants ----------
#define BB 16
#define NN 128
#define DE 4
#define DX 11
#define HH 128
#define LL 127   // eigen outputs after dropping smallest
#define LAMBDA_BOUND 1.0e-4f
#define NSWEEP 10
#define LDP 129  // padded LDS row stride (kills 32-way bank conflicts on column access)

typedef float v2f __attribute__((ext_vector_type(2)));
typedef float v8f __attribute__((ext_vector_type(8)));

// ---------- output layout (floats, concatenated in return order) ----------
#define OFF_LEMB   0                                   // B*127*H = 260096
#define OFF_LMASK  (OFF_LEMB + BB*LL*HH)               // B*127   = 2032
#define OFF_UC     (OFF_LMASK + BB*LL)                 // B*N*127*H
#define OFF_XH     (OFF_UC + BB*NN*LL*HH)              // B*N*H
#define OFF_NM     (OFF_XH + BB*NN*HH)                 // B*N

// ---------- workspace layout (floats) ----------
#define WS_LMAT    0                                   // B*N*N Laplacian / adj
#define WS_D       (WS_LMAT + BB*NN*NN)                // B*N degrees
#define WS_EVAL    (WS_D + BB*NN)                      // B*N sorted eigenvalues
#define WS_LMASKF  (WS_EVAL + BB*NN)                   // B*128 (col 127 forced masked)
#define WS_S       (WS_LMASKF + BB*NN)                 // B*128 column sums of Upad
#define WS_USRT    (WS_S + BB*NN)                      // B*N*N sorted eigenvectors
#define WS_UPAD    (WS_USRT + BB*NN*NN)                // B*N*128 masked, shifted U
#define WS_T       (WS_UPAD + BB*NN*NN)                // B*512*128  T[b, n*4+d, l]

// ---------- dynamic LDS partition for the eigh kernel (floats) ----------
#define SM_AM    0                       // 128*129 = 16512
#define SM_VM    (SM_AM + NN*LDP)        // 16512
#define SM_C     (SM_VM + NN*LDP)        // 64
#define SM_S     (SM_C + 64)             // 64
#define SM_P     (SM_S + 64)             // 64 (ints)
#define SM_Q     (SM_P + 64)             // 64 (ints)
#define SM_KEYS  (SM_Q + 64)             // 128
#define SM_KIDX  (SM_KEYS + NN)          // 128 (ints)
#define SM_TOTAL (SM_KIDX + NN)          // 33,536 floats = 134,144 bytes

// ============================================================
// K1: adjacency row -> degree; stash adj (0/1) into Lmat
// grid: B*N blocks of 128 threads (thread = m)
// ============================================================
__global__ void k_adj_deg(const float* __restrict__ A, float* __restrict__ ws) {
    __shared__ float red[NN];
    int bn = blockIdx.x;            // b*128 + n
    int m  = threadIdx.x;
    const float* a4 = A + ((size_t)bn * NN + m) * DE;
    float adjf = (a4[0] != 0.f || a4[1] != 0.f || a4[2] != 0.f || a4[3] != 0.f) ? 1.f : 0.f;
    ws[WS_LMAT + (size_t)bn * NN + m] = adjf;
    red[m] = adjf;
    __syncthreads();
    for (int s = 64; s > 0; s >>= 1) {
        if (m < s) red[m] += red[m + s];
        __syncthreads();
    }
    if (m == 0) ws[WS_D + bn] = red[0];
}

// ============================================================
// K2: L = tD_n * (D*I - adj) * tD_m   (elementwise, in place)
// ============================================================
__global__ void k_laplacian(float* __restrict__ ws) {
    int idx = blockIdx.x * blockDim.x + threadIdx.x;   // b*N*N + n*N + m
    if (idx >= BB * NN * NN) return;
    int m  = idx & (NN - 1);
    int bn = idx >> 7;
    int n  = bn & (NN - 1);
    int b  = bn >> 7;
    float adjf = ws[WS_LMAT + idx];
    float dn = ws[WS_D + b * NN + n];
    float dm = ws[WS_D + b * NN + m];
    float val = ((n == m) ? dn : 0.f) - adjf;
    val *= rsqrtf(fmaxf(dn, 1.f)) * rsqrtf(fmaxf(dm, 1.f));
    ws[WS_LMAT + idx] = val;
}

// ============================================================
// K3: batched symmetric eigh via parallel two-sided Jacobi.
// one block per batch, 1024 threads (32 wave32s on one WGP),
// ~131 KB dynamic LDS (A + V, row stride padded to 129 dwords so
// both row- and column-plane rotations are bank-conflict-free on
// the 64-bank LDS).  Uses CDNA5's 320 KB per-WGP LDS.
// ============================================================
__global__ void __launch_bounds__(1024)
k_jacobi_eigh(float* __restrict__ ws) {
    extern __shared__ float smem[];
    float* Am   = smem + SM_AM;
    float* Vm   = smem + SM_VM;
    float* cA   = smem + SM_C;
    float* sA   = smem + SM_S;
    int*   pA   = (int*)(smem + SM_P);
    int*   qA   = (int*)(smem + SM_Q);
    float* keys = smem + SM_KEYS;
    int*   kidx = (int*)(smem + SM_KIDX);

    const int b   = blockIdx.x;
    const int tid = threadIdx.x;
    const int nth = 1024;
    const float* Lg = ws + WS_LMAT + (size_t)b * NN * NN;

    for (int w = tid; w < NN * NN; w += nth) {
        int i = w >> 7, j = w & 127;
        Am[i * LDP + j] = Lg[w];
        Vm[i * LDP + j] = (i == j) ? 1.f : 0.f;
    }
    __syncthreads();

    for (int sweep = 0; sweep < NSWEEP; ++sweep) {
        for (int r = 0; r < NN - 1; ++r) {
            // 64 disjoint pairs via circle method (player 127 fixed)
            if (tid < 64) {
                int k = tid, p, q;
                if (k == 0) { p = NN - 1; q = r % (NN - 1); }
                else        { p = (r + k) % (NN - 1); q = (r - k + (NN - 1)) % (NN - 1); }
                if (p > q) { int t = p; p = q; q = t; }
                float app = Am[p * LDP + p];
                float aqq = Am[q * LDP + q];
                float apq = Am[p * LDP + q];
                float c = 1.f, s = 0.f;
                if (fabsf(apq) > 1e-20f) {
                    float tau = (aqq - app) / (2.f * apq);
                    float t   = ((tau >= 0.f) ? 1.f : -1.f) / (fabsf(tau) + sqrtf(1.f + tau * tau));
                    c = rsqrtf(1.f + t * t);
                    s = t * c;
                }
                cA[tid] = c; sA[tid] = s; pA[tid] = p; qA[tid] = q;
            }
            __syncthreads();
            // row rotations: rows p,q of A (consecutive j per wave: conflict-free)
            for (int w = tid; w < 64 * NN; w += nth) {
                int k = w >> 7, j = w & 127;
                int p = pA[k], q = qA[k];
                float c = cA[k], s = sA[k];
                float apj = Am[p * LDP + j], aqj = Am[q * LDP + j];
                Am[p * LDP + j] = c * apj - s * aqj;
                Am[q * LDP + j] = s * apj + c * aqj;
            }
            __syncthreads();
            // column rotations: cols p,q of A and V
            // (stride-129 rows => bank = (i + p) mod 64: conflict-free)
            for (int w = tid; w < 64 * NN; w += nth) {
                int k = w >> 7, i = w & 127;
                int p = pA[k], q = qA[k];
                float c = cA[k], s = sA[k];
                float aip = Am[i * LDP + p], aiq = Am[i * LDP + q];
                Am[i * LDP + p] = c * aip - s * aiq;
                Am[i * LDP + q] = s * aip + c * aiq;
                float vip = Vm[i * LDP + p], viq = Vm[i * LDP + q];
                Vm[i * LDP + p] = c * vip - s * viq;
                Vm[i * LDP + q] = s * vip + c * viq;
            }
            __syncthreads();
        }
    }

    // sort eigenvalues ascending (odd-even transposition), track permutation
    if (tid < NN) { keys[tid] = Am[tid * LDP + tid]; kidx[tid] = tid; }
    __syncthreads();
    for (int it = 0; it < NN; ++it) {
        if (tid < 64) {
            int j = 2 * tid + (it & 1);
            if (j + 1 < NN && keys[j] > keys[j + 1]) {
                float tk = keys[j]; keys[j] = keys[j + 1]; keys[j + 1] = tk;
                int   ti = kidx[j]; kidx[j] = kidx[j + 1]; kidx[j + 1] = ti;
            }
        }
        __syncthreads();
    }
    if (tid < NN) ws[WS_EVAL + b * NN + tid] = keys[tid];
    for (int w = tid; w < NN * NN; w += nth) {
        int i = w >> 7, j = w & 127;
        ws[WS_USRT + (size_t)b * NN * NN + i * NN + j] = Vm[i * LDP + kidx[j]];
    }
}

// ============================================================
// K4: lambda outputs. grid: B blocks, 128 threads (thread = l or pad)
// ============================================================
__global__ void k_lambda(const float* __restrict__ mu, const float* __restrict__ inv_sigma,
                         float* __restrict__ ws, float* __restrict__ out) {
    int b = blockIdx.x, j = threadIdx.x;
    if (j < LL) {
        float lam  = ws[WS_EVAL + b * NN + j + 1];
        int   msk  = (fabsf(lam) < LAMBDA_BOUND) ? 1 : 0;
        ws[WS_LMASKF + b * NN + j] = (float)msk;
        out[OFF_LMASK + b * LL + j] = (float)msk;
        float sl = sqrtf(fmaxf(lam, 0.f));
        for (int h = 0; h < HH; ++h) {
            float d = (sl - mu[h]) * inv_sigma[h];
            out[OFF_LEMB + ((size_t)b * LL + j) * HH + h] = __expf(-d * d);
        }
    } else {
        ws[WS_LMASKF + b * NN + j] = 1.f;   // force pad column 127 to zero
    }
}

// ============================================================
// K5: Upad[b,i,l] = masked, shifted eigenvectors (l in 0..127, col 127 = 0)
// grid: B*N blocks of 128 threads; also emits nodemask output
// ============================================================
__global__ void k_upad(const unsigned char* __restrict__ nodemask,
                       float* __restrict__ ws, float* __restrict__ out) {
    int bn = blockIdx.x;                   // b*128 + i
    int b  = bn >> 7;
    int lp = threadIdx.x;
    unsigned char nm = nodemask[bn];
    float val = 0.f;
    if (lp < LL) {
        val = ws[WS_USRT + (size_t)bn * NN + (lp + 1)];
        if (ws[WS_LMASKF + b * NN + lp] != 0.f || nm) val = 0.f;
    }
    ws[WS_UPAD + (size_t)bn * NN + lp] = val;
    if (lp == 0) out[OFF_NM + bn] = nm ? 1.f : 0.f;
}

// ============================================================
// K6: S[b,l] = sum_i Upad[b,i,l].  grid: B blocks, 128 threads
// ============================================================
__global__ void k_colsum(float* __restrict__ ws) {
    int b = blockIdx.x, lp = threadIdx.x;
    float s = 0.f;
    const float* U = ws + WS_UPAD + (size_t)b * NN * NN;
    for (int i = 0; i < NN; ++i) s += U[i * NN + lp];
    ws[WS_S + b * NN + lp] = s;
}

// ============================================================
// K7: T[b, n*4+d, l] = sum_m A[b,n,m,d] * Upad[b,m,l]
// WMMA f32 16x16x4, K=128 -> 32 chained wmma per 16x16 tile.
// one wave per tile; tiles per batch = (512/16)*(128/16) = 256.
// grid: 1024 blocks of 128 threads (4 waves).
// ============================================================
__global__ void k_T_wmma(const float* __restrict__ A, float* __restrict__ ws) {
    int wave = blockIdx.x * 4 + (threadIdx.x >> 5);
    int lane = threadIdx.x & 31;
    int b  = wave >> 8;
    int t  = wave & 255;
    int mt = t >> 3;            // 0..31  (rows of 16 over 512)
    int nt = t & 7;             // 0..7   (cols of 16 over 128)
    int R0 = mt * 16, C0 = nt * 16;

    int row = lane & 15;
    int hi  = lane >> 4;        // 0 or 1
    int kb  = hi * 2;           // K offset within k-quad: {0,2}
    int col = lane & 15;

    int gRow = R0 + row;        // = n*4 + d
    int n = gRow >> 2, d = gRow & 3;
    const float* Abase = A + (((size_t)b * NN + n) * NN) * DE + d;  // + m*DE
    const float* Ub    = ws + WS_UPAD + (size_t)b * NN * NN;

    v8f acc = {0.f, 0.f, 0.f, 0.f, 0.f, 0.f, 0.f, 0.f};
    for (int kk = 0; kk < NN / 4; ++kk) {
        int k0 = kk * 4;
        // speculative prefetch of A 16 k ahead (four k-quads) (global_prefetch;
        // OOB speculative prefetches are dropped silently per ISA)
        __builtin_prefetch(Abase + (size_t)(k0 + 16) * DE, 0, 1);
        v2f av, bv;
        av.x = Abase[(size_t)(k0 + kb) * DE];
        av.y = Abase[(size_t)(k0 + kb + 1) * DE];
        bv.x = Ub[(k0 + kb) * NN + C0 + col];
        bv.y = Ub[(k0 + kb + 1) * NN + C0 + col];
        acc = __builtin_amdgcn_wmma_f32_16x16x4_f32(false, av, false, bv,
                                                    (short)0, acc, false, false);
    }
    float* T = ws + WS_T + (size_t)b * 512 * NN;
    for (int r = 0; r < 8; ++r) {
        int M = r + hi * 8;
        T[(size_t)(R0 + M) * NN + C0 + col] = acc[r];
    }
}

// ============================================================
// K8: Uc[b,n,l,h] = (1-mask_n)*( sum_d T[b,n,l,d]*W_edge[d,h] + b_edge[h]*S[b,l] )
// K = DE = 4 -> a single wmma_f32_16x16x4 per 16x16 tile.
// one wave per (b,n,l-tile); loops 8 h-tiles reusing the A operand.
// grid: 4096 blocks of 128 threads.
// ============================================================
__global__ void k_Uc_wmma(const float* __restrict__ W_edge, const float* __restrict__ b_edge,
                          const unsigned char* __restrict__ nodemask,
                          const float* __restrict__ ws, float* __restrict__ out) {
    int wave = blockIdx.x * 4 + (threadIdx.x >> 5);
    int lane = threadIdx.x & 31;
    int b   = wave >> 10;
    int rem = wave & 1023;
    int n   = rem >> 3;
    int lt  = rem & 7;
    int lp0 = lt * 16;

    int row = lane & 15;
    int hi  = lane >> 4;
    int kb  = hi * 2;
    int col = lane & 15;

    const float* T = ws + WS_T + (size_t)b * 512 * NN + (size_t)n * 4 * NN;
    v2f av;
    av.x = T[(size_t)kb * NN + lp0 + row];
    av.y = T[(size_t)(kb + 1) * NN + lp0 + row];

    float sv[8];
    for (int r = 0; r < 8; ++r) sv[r] = ws[WS_S + b * NN + lp0 + r + hi * 8];
    float nmf = nodemask[b * NN + n] ? 0.f : 1.f;

    for (int ht = 0; ht < 8; ++ht) {
        int h0 = ht * 16;
        v2f bv;
        bv.x = W_edge[kb * HH + h0 + col];
        bv.y = W_edge[(kb + 1) * HH + h0 + col];
        float be = b_edge[h0 + col];
        v8f c;
        for (int r = 0; r < 8; ++r) c[r] = be * sv[r];
        c = __builtin_amdgcn_wmma_f32_16x16x4_f32(false, av, false, bv,
                                                  (short)0, c, false, false);
        for (int r = 0; r < 8; ++r) {
            int lp = lp0 + r + hi * 8;
            if (lp < LL)
                out[OFF_UC + (((size_t)b * NN + n) * LL + lp) * HH + h0 + col] = nmf * c[r];
        }
    }
}

// ============================================================
// K9: Xh = X @ W_x + b_x, node-masked. One thread per (b,n,h).
// ============================================================
__global__ void k_xh(const float* __restrict__ X, const float* __restrict__ W_x,
                     const float* __restrict__ b_x, const unsigned char* __restrict__ nodemask,
                     float* __restrict__ out) {
    int g = blockIdx.x * blockDim.x + threadIdx.x;
    if (g >= BB * NN * HH) return;
    int h  = g & (HH - 1);
    int bn = g >> 7;
    float acc = b_x[h];
    const float* x = X + (size_t)bn * DX;
    for (int d = 0; d < DX; ++d) acc += x[d] * W_x[d * HH + h];
    out[OFF_XH + g] = nodemask[bn] ? 0.f : acc;
}

// ============================================================
extern "C" void kernel_launch(void* const* d_in, const int* in_sizes, int n_in,
                              void* d_out, int out_size, void* d_ws, size_t ws_size,
                              hipStream_t stream) {
    const float* A          = (const float*)d_in[0];
    const float* X          = (const float*)d_in[1];
    const unsigned char* nm = (const unsigned char*)d_in[2];
    const float* W_edge     = (const float*)d_in[3];
    const float* b_edge     = (const float*)d_in[4];
    const float* W_x        = (const float*)d_in[5];
    const float* b_x        = (const float*)d_in[6];
    const float* mu         = (const float*)d_in[7];
    const float* inv_sigma  = (const float*)d_in[8];
    float* out = (float*)d_out;
    float* ws  = (float*)d_ws;

    (void)in_sizes; (void)n_in; (void)out_size; (void)ws_size;

    // 1) adjacency + degrees, then Laplacian
    k_adj_deg<<<BB * NN, NN, 0, stream>>>(A, ws);
    k_laplacian<<<(BB * NN * NN) / 256, 256, 0, stream>>>(ws);

    // 2) batched Jacobi eigh (one WGP per batch, 32 waves, ~131 KB LDS)
    size_t lds = (size_t)SM_TOTAL * sizeof(float);
    k_jacobi_eigh<<<BB, 1024, lds, stream>>>(ws);

    // 3) lambda embedding / masks, masked padded U, column sums
    k_lambda<<<BB, NN, 0, stream>>>(mu, inv_sigma, ws, out);
    k_upad<<<BB * NN, NN, 0, stream>>>(nm, ws, out);
    k_colsum<<<BB, NN, 0, stream>>>(ws);

    // 4) T = contract(A, U)  -- fp32 WMMA, K=128
    k_T_wmma<<<(BB * 256) / 4, 128, 0, stream>>>(A, ws);

    // 5) Uc = T @ W_edge + b_edge (x) S  -- fp32 WMMA, K=4
    k_Uc_wmma<<<(BB * NN * 8) / 4, 128, 0, stream>>>(W_edge, b_edge, nm, ws, out);

    // 6) Xh
    k_xh<<<(BB * NN * HH) / 256, 256, 0, stream>>>(X, W_x, b_x, nm, out);
}